// _ECELoss_87780541595820
// MI455X (gfx1250) — compile-verified
//
#include <hip/hip_runtime.h>
#include <hip/hip_bf16.h>
#include <math.h>

#define N_BINS 25
#define NCOLS  100                 // fixed by the reference
#define TROWS  8                   // rows per tile = waves per block
#define TILE_BYTES   (TROWS * NCOLS * 4)      // 3200 B of real data
#define CHUNKS       (TILE_BYTES / 16)        // 200 x 16B chunks
#define BUF_FLOATS   1024                     // 4096 B buffer (dummy slots for tid>=200)

typedef __attribute__((ext_vector_type(4))) float fvec4;

// ---- gfx1250 async global->LDS copy (ASYNCcnt path) ------------------------
// Probe round 2 showed the builtin exists and takes pointers to 16B int
// vectors: global side addrspace(1), LDS side addrspace(3).
typedef int v4i __attribute__((vector_size(16)));
typedef __attribute__((address_space(1))) v4i gv4i;
typedef __attribute__((address_space(3))) v4i lv4i;

#if __has_builtin(__builtin_amdgcn_global_load_async_to_lds_b128)
#define ASYNC_LOAD16(gp, lp)                                                   \
    __builtin_amdgcn_global_load_async_to_lds_b128(                            \
        (gv4i*)(gp), (lv4i*)(lp), 0, 0)
#else
#define ASYNC_LOAD16(gp, lp)                                                   \
    do {                                                                       \
        unsigned __lo = (unsigned)(unsigned long long)(lv4i*)(lp);             \
        unsigned long long __ga = (unsigned long long)(gp);                    \
        asm volatile("global_load_async_to_lds_b128 %0, %1, off"               \
                     :: "v"(__lo), "v"(__ga) : "memory");                      \
    } while (0)
#endif

#if __has_builtin(__builtin_amdgcn_s_wait_asynccnt)
#define WAIT_ASYNC(n) __builtin_amdgcn_s_wait_asynccnt(n)
#else
#define WAIT_ASYNC(n) asm volatile("s_wait_asynccnt %0" :: "i"(n))
#endif

// ---------------------------------------------------------------------------
// Kernel 0: zero the 75-float bin workspace (harness poisons d_ws with 0xAA).
// ---------------------------------------------------------------------------
__global__ void ece_zero_ws(float* __restrict__ ws) {
    int t = blockIdx.x * blockDim.x + threadIdx.x;
    if (t < 3 * N_BINS) ws[t] = 0.0f;
}

// ---------------------------------------------------------------------------
// Kernel 1: streaming pass with a double-buffered async global->LDS pipeline.
// Each block grid-strides over 8-row tiles (3200 B). All 256 threads issue one
// global_load_async_to_lds_b128 (16 B/lane) for tile k+1 while the 8 waves
// reduce tile k out of LDS (ds_load_b128). s_wait_asynccnt + s_barrier
// synchronize the two stages. Per row: wave32 butterfly max/argmax, then
// sum(exp(x-max)); conf = 1/sum. Bins accumulate via ds_add_f32, flushed with
// 75 global_atomic_add_f32 per block.
// ---------------------------------------------------------------------------
__global__ __launch_bounds__(256) void ece_partial(
        const float* __restrict__ logits,
        const int*   __restrict__ labels,
        float*       __restrict__ ws,
        int n_rows) {
    __shared__ __align__(16) float s_stage[2][BUF_FLOATS];
    __shared__ float s_cnt[N_BINS];
    __shared__ float s_conf[N_BINS];
    __shared__ float s_acc[N_BINS];

    const int tid = threadIdx.x;
    if (tid < N_BINS) { s_cnt[tid] = 0.0f; s_conf[tid] = 0.0f; s_acc[tid] = 0.0f; }
    __syncthreads();

    const int lane = tid & 31;
    const int wv   = tid >> 5;                    // wave id in block = row in tile
    const bool act = lane < (NCOLS / 4);          // lanes 0..24 carry row data

    const int nt   = (n_rows + TROWS - 1) / TROWS;      // #tiles
    const int step = gridDim.x;
    const long long lastChunkByte = (long long)n_rows * NCOLS * 4 - 16;

    // chunk this thread copies (threads >=200 duplicate last chunk into
    // private dummy LDS slots: same cacheline, keeps EXEC all-ones and
    // per-wave ASYNCcnt uniform)
    const int myChunk = tid < CHUNKS ? tid : (CHUNKS - 1);

    int tile = blockIdx.x;
    int buf  = 0;

    // Prologue: start fill of first tile.
    if (tile < nt) {
        long long g = (long long)tile * TILE_BYTES + (long long)myChunk * 16;
        if (g > lastChunkByte) g = lastChunkByte;
        ASYNC_LOAD16((const char*)logits + g, &s_stage[0][tid * 4]);
    }

    for (; tile < nt; tile += step) {
        const int next = tile + step;

        // Stage fill for the next tile into the other buffer.
        if (next < nt) {
            long long g = (long long)next * TILE_BYTES + (long long)myChunk * 16;
            if (g > lastChunkByte) g = lastChunkByte;
            ASYNC_LOAD16((const char*)logits + g, &s_stage[buf ^ 1][tid * 4]);
            WAIT_ASYNC(1);     // in-order: current tile's fill has completed
        } else {
            WAIT_ASYNC(0);     // drain final fill
        }
        __syncthreads();       // all waves' chunks of this tile visible in LDS

        // ---- consume: wave wv reduces row (tile*8 + wv) from LDS ----------
        const int row = tile * TROWS + wv;
        if (row < n_rows) {
            fvec4 v = *(const fvec4*)&s_stage[buf][wv * NCOLS + (act ? lane * 4 : 0)];

            // per-lane max + first-occurrence argmax over 4 elements
            float m; int mi;
            if (act) {
                int base = lane * 4;
                m = v.x; mi = base;
                if (v.y > m) { m = v.y; mi = base + 1; }
                if (v.z > m) { m = v.z; mi = base + 2; }
                if (v.w > m) { m = v.w; mi = base + 3; }
            } else {
                m = -__builtin_inff(); mi = 0x7FFFFFFF;
            }
            // wave32 butterfly reduce: max with lowest-index tie-break
            #pragma unroll
            for (int off = 16; off > 0; off >>= 1) {
                float om = __shfl_xor(m, off, 32);
                int   oi = __shfl_xor(mi, off, 32);
                if (om > m || (om == m && oi < mi)) { m = om; mi = oi; }
            }
            // sum of exp(x - max); max softmax prob = 1/sum
            float s = 0.0f;
            if (act) {
                s = __expf(v.x - m) + __expf(v.y - m) +
                    __expf(v.z - m) + __expf(v.w - m);
            }
            #pragma unroll
            for (int off = 16; off > 0; off >>= 1)
                s += __shfl_xor(s, off, 32);

            if (lane == 0) {
                float conf = 1.0f / s;                        // in (0,1]
                int bin = (int)ceilf(conf * (float)N_BINS) - 1;
                bin = bin < 0 ? 0 : (bin > N_BINS - 1 ? N_BINS - 1 : bin);
                float acc = (mi == labels[row]) ? 1.0f : 0.0f;
                atomicAdd(&s_cnt[bin], 1.0f);                 // ds_add_f32
                atomicAdd(&s_conf[bin], conf);
                atomicAdd(&s_acc[bin], acc);
            }
        }

        __syncthreads();       // everyone done reading buf before it refills
        buf ^= 1;
    }

    __syncthreads();
    if (tid < N_BINS) {        // 75 global f32 atomics per block
        unsafeAtomicAdd(&ws[tid],              s_cnt[tid]);
        unsafeAtomicAdd(&ws[N_BINS + tid],     s_conf[tid]);
        unsafeAtomicAdd(&ws[2 * N_BINS + tid], s_acc[tid]);
    }
}

// ---------------------------------------------------------------------------
// Kernel 2: one wave folds 25 bins into the scalar ECE.
// ---------------------------------------------------------------------------
__global__ void ece_final(const float* __restrict__ ws,
                          float* __restrict__ out, int n_rows) {
    int lane = threadIdx.x;
    float term = 0.0f;
    if (lane < N_BINS) {
        float cnt   = ws[lane];
        float sconf = ws[N_BINS + lane];
        float sacc  = ws[2 * N_BINS + lane];
        if (cnt > 0.0f) {
            float inv = 1.0f / cnt;
            term = fabsf(sconf * inv - sacc * inv) * (cnt / (float)n_rows);
        }
    }
    #pragma unroll
    for (int off = 16; off > 0; off >>= 1)
        term += __shfl_xor(term, off, 32);
    if (lane == 0) out[0] = term;
}

// ---------------------------------------------------------------------------
extern "C" void kernel_launch(void* const* d_in, const int* in_sizes, int n_in,
                              void* d_out, int out_size, void* d_ws, size_t ws_size,
                              hipStream_t stream) {
    const float* logits = (const float*)d_in[0];
    const int*   labels = (const int*)d_in[1];
    float* out = (float*)d_out;
    float* ws  = (float*)d_ws;
    const int n_rows = in_sizes[1];     // 500000 labels -> row count

    ece_zero_ws<<<1, 128, 0, stream>>>(ws);

    // 2048 blocks x 8 waves, each block pipelining 8-row tiles through LDS:
    // enough in-flight HBM traffic to saturate 23.3 TB/s, tiny atomic tail.
    ece_partial<<<2048, 256, 0, stream>>>(logits, labels, ws, n_rows);

    ece_final<<<1, 32, 0, stream>>>(ws, out, n_rows);
}